// AdaptativeGCN_35545149341868
// MI455X (gfx1250) — compile-verified
//
#include <hip/hip_runtime.h>

typedef __attribute__((ext_vector_type(2))) float v2f;
typedef __attribute__((ext_vector_type(8))) float v8f;

#define KSPLIT 16

// ---------------------------------------------------------------------------
// Zero the WMMA accumulator buffers (must run every call: we atomicAdd later).
// ---------------------------------------------------------------------------
__global__ void zero_kernel(float* __restrict__ p, int count) {
  int i = blockIdx.x * blockDim.x + threadIdx.x;
  if (i < count) p[i] = 0.0f;
}

// ---------------------------------------------------------------------------
// Dense pre-projections (tiny GEMMs, VALU is fine):
//   U1  = X @ Ws1                       [N,32]  (A-side of secondary conv)
//   Z1  = X @ Ws2 + bs                  [N,32]  (skip term of secondary conv)
//   P23[:,7:14] = cat(X,path) @ Wma     [N,7]   (A-side of main conv)
//   Z3  = cat(X,path) @ Wmb + bm        [N,7]
//   P23[:,14:16] = 0 (pad)
// One thread per (row, j), j in [0,80).
// ---------------------------------------------------------------------------
__global__ void proj1_kernel(const float* __restrict__ X,
                             const float* __restrict__ P,
                             const float* __restrict__ Ws1,
                             const float* __restrict__ Ws2,
                             const float* __restrict__ bs,
                             const float* __restrict__ Wma,
                             const float* __restrict__ Wmb,
                             const float* __restrict__ bm,
                             float* __restrict__ U1,
                             float* __restrict__ Z1,
                             float* __restrict__ P23,
                             float* __restrict__ Z3,
                             int n) {
  int idx = blockIdx.x * blockDim.x + threadIdx.x;
  int row = idx / 80;
  int j = idx - row * 80;
  if (row >= n) return;
  const float* x  = X + (size_t)row * 128;
  const float* pr = P + (size_t)row * 64;
  if (j < 32) {
    float s = 0.f;
    for (int k = 0; k < 128; ++k) s = fmaf(x[k], Ws1[k * 32 + j], s);
    U1[(size_t)row * 32 + j] = s;
  } else if (j < 64) {
    int c = j - 32;
    float s = bs[c];
    for (int k = 0; k < 128; ++k) s = fmaf(x[k], Ws2[k * 32 + c], s);
    Z1[(size_t)row * 32 + c] = s;
  } else if (j < 71) {
    int c = j - 64;
    float s = 0.f;
    for (int k = 0; k < 128; ++k) s = fmaf(x[k], Wma[k * 7 + c], s);
    for (int k = 0; k < 64; ++k)  s = fmaf(pr[k], Wma[(128 + k) * 7 + c], s);
    P23[(size_t)row * 16 + 7 + c] = s;
  } else if (j < 78) {
    int c = j - 71;
    float s = bm[c];
    for (int k = 0; k < 128; ++k) s = fmaf(x[k], Wmb[k * 7 + c], s);
    for (int k = 0; k < 64; ++k)  s = fmaf(pr[k], Wmb[(128 + k) * 7 + c], s);
    Z3[(size_t)row * 7 + c] = s;
  } else {
    P23[(size_t)row * 16 + 14 + (j - 78)] = 0.f;  // pad cols 14,15
  }
}

// ---------------------------------------------------------------------------
// The big pass: S += A @ B over a K-chunk, using V_WMMA_F32_16X16X4_F32.
// B is [n, COLTILES*16] row-major. Each wave owns one 16-row tile and
// COLTILES 16-col tiles, accumulating its K-chunk in registers, then flushes
// with global_atomic_add_f32 (split-K reduction).
//
// Software-pipelined: loads for step i+1 are issued before the WMMAs that
// consume step i, so s_wait_loadcnt overlaps matrix math with VMEM latency.
// A loads are non-temporal (each 128B line of A is dead after an 8-step
// window; keep L2 for B and the small per-row buffers).
//
// Fragment layouts (ISA 7.12.2, 32-bit):
//   A 16x4 : lane<16 -> {K=k,k+1}, lane>=16 -> {K=k+2,k+3}, M = lane&15
//   B 4x16 : VGPR v, lanes<16 -> row K=k+v, lanes>=16 -> row K=k+2+v, N=lane&15
//   C 16x16: VGPR i, lanes<16 -> M=i, lanes>=16 -> M=i+8, N=lane&15
// Wave-uniform control flow keeps EXEC all-ones across WMMA as required.
// ---------------------------------------------------------------------------
template <int COLTILES>
__global__ void spmm_wmma_kernel(const float* __restrict__ A,
                                 const float* __restrict__ B,
                                 float* __restrict__ S,
                                 int n, int kChunk) {
  const int ldb = COLTILES * 16;
  int wave = threadIdx.x >> 5;
  int lane = threadIdx.x & 31;
  int rowTile = blockIdx.x * 8 + wave;
  int numRowTiles = n >> 4;
  if (rowTile >= numRowTiles) return;  // wave-uniform exit

  int m0 = rowTile << 4;
  int k0 = blockIdx.y * kChunk;
  int k1 = k0 + kChunk; if (k1 > n) k1 = n;   // (k1-k0) stays multiple of 4

  int hi = lane >> 4;      // 0: K pair {0,1}, 1: K pair {2,3}
  int ml = lane & 15;      // M (A) / N (B,C)

  const float* aPtr = A + (size_t)(m0 + ml) * n + (k0 + 2 * hi);
  const float* bPtr = B + (size_t)(k0 + 2 * hi) * ldb + ml;

  v8f acc[COLTILES] = {};

  // ---- pipeline prologue: fragments for k0 ----
  v2f a, b[COLTILES];
  a.x = __builtin_nontemporal_load(aPtr + 0);
  a.y = __builtin_nontemporal_load(aPtr + 1);
#pragma unroll
  for (int t = 0; t < COLTILES; ++t) {
    b[t].x = bPtr[t * 16];
    b[t].y = bPtr[ldb + t * 16];
  }

  // ---- steady state: load k, consume k-4 ----
  for (int k = k0 + 4; k < k1; k += 4) {
    aPtr += 4;
    bPtr += 4 * (size_t)ldb;
    v2f an, bn[COLTILES];
    an.x = __builtin_nontemporal_load(aPtr + 0);
    an.y = __builtin_nontemporal_load(aPtr + 1);
#pragma unroll
    for (int t = 0; t < COLTILES; ++t) {
      bn[t].x = bPtr[t * 16];
      bn[t].y = bPtr[ldb + t * 16];
    }
#pragma unroll
    for (int t = 0; t < COLTILES; ++t) {
      acc[t] = __builtin_amdgcn_wmma_f32_16x16x4_f32(
          false, a, false, b[t], (short)0, acc[t], false, false);
    }
    a = an;
#pragma unroll
    for (int t = 0; t < COLTILES; ++t) b[t] = bn[t];
  }

  // ---- epilogue: consume final fragments ----
#pragma unroll
  for (int t = 0; t < COLTILES; ++t) {
    acc[t] = __builtin_amdgcn_wmma_f32_16x16x4_f32(
        false, a, false, b[t], (short)0, acc[t], false, false);
  }

  // ---- split-K flush ----
#pragma unroll
  for (int t = 0; t < COLTILES; ++t) {
#pragma unroll
    for (int i = 0; i < 8; ++i) {
      int row = m0 + i + 8 * hi;
      int col = t * 16 + ml;
      atomicAdd(&S[(size_t)row * ldb + col], acc[t][i]);
    }
  }
}

// ---------------------------------------------------------------------------
// sec = relu(S1 + Z1); sec_cat = [sec, path];
//   P23[:,0:7] = sec_cat @ Wm2a ; Z2 = sec_cat @ Wm2b + bm2
// One thread per (row, j), j in [0,14): j<7 -> Wm2a column, else Wm2b column.
// ---------------------------------------------------------------------------
__global__ void proj2_kernel(const float* __restrict__ S1,
                             const float* __restrict__ Z1,
                             const float* __restrict__ Ppath,
                             const float* __restrict__ Wm2a,
                             const float* __restrict__ Wm2b,
                             const float* __restrict__ bm2,
                             float* __restrict__ P23,
                             float* __restrict__ Z2,
                             int n) {
  int idx = blockIdx.x * blockDim.x + threadIdx.x;
  int row = idx / 14;
  int j = idx - row * 14;
  if (row >= n) return;
  bool isA = (j < 7);
  int c = isA ? j : j - 7;
  const float* W = isA ? Wm2a : Wm2b;
  float s = isA ? 0.f : bm2[c];
  const float* s1 = S1 + (size_t)row * 32;
  const float* z1 = Z1 + (size_t)row * 32;
  for (int k = 0; k < 32; ++k) {
    float sec = fmaxf(s1[k] + z1[k], 0.f);
    s = fmaf(sec, W[k * 7 + c], s);
  }
  const float* p = Ppath + (size_t)row * 64;
  for (int k = 0; k < 64; ++k) s = fmaf(p[k], W[(32 + k) * 7 + c], s);
  if (isA) P23[(size_t)row * 16 + c] = s;
  else     Z2[(size_t)row * 7 + c] = s;
}

// ---------------------------------------------------------------------------
// main_sec = S2[:,0:7]+Z2; main = S2[:,7:14]+Z3;
// combined = 0.5*(v2*main + v1*main_sec); softmax over 7. One thread per row.
// ---------------------------------------------------------------------------
__global__ void final_kernel(const float* __restrict__ S2,
                             const float* __restrict__ Z2,
                             const float* __restrict__ Z3,
                             const float* __restrict__ v1p,
                             const float* __restrict__ v2p,
                             float* __restrict__ out,
                             int n) {
  int row = blockIdx.x * blockDim.x + threadIdx.x;
  if (row >= n) return;
  float v1 = v1p[0], v2 = v2p[0];
  const float* s2 = S2 + (size_t)row * 16;
  const float* z2 = Z2 + (size_t)row * 7;
  const float* z3 = Z3 + (size_t)row * 7;
  float c[7];
  float mx = -3.4e38f;
  for (int j = 0; j < 7; ++j) {
    float ms = s2[j] + z2[j];
    float mn = s2[7 + j] + z3[j];
    c[j] = 0.5f * (v2 * mn + v1 * ms);
    mx = fmaxf(mx, c[j]);
  }
  float sum = 0.f;
  for (int j = 0; j < 7; ++j) { c[j] = expf(c[j] - mx); sum += c[j]; }
  float inv = 1.f / sum;
  float* o = out + (size_t)row * 7;
  for (int j = 0; j < 7; ++j) o[j] = c[j] * inv;
}

// ---------------------------------------------------------------------------
extern "C" void kernel_launch(void* const* d_in, const int* in_sizes, int n_in,
                              void* d_out, int out_size, void* d_ws, size_t ws_size,
                              hipStream_t stream) {
  const float* X    = (const float*)d_in[0];   // [N,128]
  const float* A    = (const float*)d_in[1];   // [N,N]
  const float* P    = (const float*)d_in[2];   // [N,64]
  const float* Ws1  = (const float*)d_in[3];   // [128,32]
  const float* Ws2  = (const float*)d_in[4];
  const float* bs   = (const float*)d_in[5];
  const float* Wm2a = (const float*)d_in[6];   // [96,7]
  const float* Wm2b = (const float*)d_in[7];
  const float* bm2  = (const float*)d_in[8];
  const float* Wma  = (const float*)d_in[9];   // [192,7]
  const float* Wmb  = (const float*)d_in[10];
  const float* bm   = (const float*)d_in[11];
  const float* v1   = (const float*)d_in[12];
  const float* v2   = (const float*)d_in[13];
  float* out = (float*)d_out;

  int n = in_sizes[0] / 128;                   // 10000 (multiple of 16)

  float* ws = (float*)d_ws;
  size_t o = 0;
  float* U1  = ws + o; o += (size_t)n * 32;    // X @ Ws1
  float* Z1  = ws + o; o += (size_t)n * 32;    // X @ Ws2 + bs
  float* S1  = ws + o; o += (size_t)n * 32;    // A @ U1       (atomic acc)
  float* S2  = ws + o; o += (size_t)n * 16;    // A @ P23      (atomic acc)
  float* P23 = ws + o; o += (size_t)n * 16;    // [P2 | P3 | pad]
  float* Z2  = ws + o; o += (size_t)n * 7;
  float* Z3  = ws + o; o += (size_t)n * 7;

  int kChunk = ((n + KSPLIT - 1) / KSPLIT + 3) & ~3;   // multiple of 4
  int rowTiles = n / 16;
  dim3 spmmGrid((rowTiles + 7) / 8, KSPLIT);

  // 1) zero the S1,S2 accumulators (contiguous region of n*48 floats)
  {
    int cnt = n * 48;
    zero_kernel<<<(cnt + 255) / 256, 256, 0, stream>>>(S1, cnt);
  }
  // 2) small dense projections
  {
    int cnt = n * 80;
    proj1_kernel<<<(cnt + 255) / 256, 256, 0, stream>>>(
        X, P, Ws1, Ws2, bs, Wma, Wmb, bm, U1, Z1, P23, Z3, n);
  }
  // 3) pass 1 over A: S1 = A @ U1   (16x32 tile per wave, 2 WMMAs/step)
  spmm_wmma_kernel<2><<<spmmGrid, 256, 0, stream>>>(A, U1, S1, n, kChunk);
  // 4) relu+concat+second projection
  {
    int cnt = n * 14;
    proj2_kernel<<<(cnt + 255) / 256, 256, 0, stream>>>(
        S1, Z1, P, Wm2a, Wm2b, bm2, P23, Z2, n);
  }
  // 5) pass 2 over A: S2 = A @ P23  (both 7-wide convs fused into 16 cols)
  spmm_wmma_kernel<1><<<spmmGrid, 256, 0, stream>>>(A, P23, S2, n, kChunk);
  // 6) combine + softmax
  final_kernel<<<(n + 255) / 256, 256, 0, stream>>>(S2, Z2, Z3, v1, v2, out, n);
}